// GCN_52484500357663
// MI455X (gfx1250) — compile-verified
//
#include <hip/hip_runtime.h>
#include <hip/hip_bf16.h>

typedef __attribute__((ext_vector_type(2))) float v2f;
typedef __attribute__((ext_vector_type(8))) float v8f;

// ---------------------------------------------------------------------------
// Degree / normalization
// ---------------------------------------------------------------------------
__global__ void deg_init_kernel(float* __restrict__ deg, int n) {
    int i = blockIdx.x * blockDim.x + threadIdx.x;
    if (i < n) deg[i] = 1.0f;   // self-loop contributes 1
}

__global__ void deg_edge_kernel(const int* __restrict__ dst, float* __restrict__ deg, int e) {
    int i = blockIdx.x * blockDim.x + threadIdx.x;
    if (i < e) atomicAdd(&deg[dst[i]], 1.0f);
}

__global__ void deg_finish_kernel(float* __restrict__ deg, int n) {
    int i = blockIdx.x * blockDim.x + threadIdx.x;
    if (i < n) deg[i] = rsqrtf(deg[i]);   // deg >= 1 always (self-loop)
}

// ---------------------------------------------------------------------------
// Tiled fp32 GEMM on the WMMA pipe: C[M,N] = (relu?)A[M,K] x B[K,N]
// block = 256 threads = 8 waves; each wave owns a 16x16 tile; a block covers
// 16 rows x 128 cols; A tile staged once in LDS (float4) for all 8 waves.
// Full-tile fast paths avoid per-element exec-mask guards in prologue and
// epilogue (stores become base + compile-time immediate offsets).
// ---------------------------------------------------------------------------
template <int K, int N, bool RELU>
__global__ void gemm16_wmma_kernel(const float* __restrict__ A,
                                   const float* __restrict__ B,
                                   float* __restrict__ C, int M) {
    __shared__ float sA[16 * K];            // [16][K] row-major
    const int m0  = blockIdx.x * 16;
    const int tid = threadIdx.x;
    const bool full = (m0 + 16 <= M);

    constexpr int KV = K / 4;               // float4s per row
    if (full) {
        for (int idx = tid; idx < 16 * KV; idx += 256) {
            const int r  = idx / KV;        // power of two -> shift
            const int kv = (idx & (KV - 1)) * 4;
            float4 v = *(const float4*)(A + (size_t)(m0 + r) * K + kv);
            if (RELU) {
                v.x = fmaxf(v.x, 0.0f); v.y = fmaxf(v.y, 0.0f);
                v.z = fmaxf(v.z, 0.0f); v.w = fmaxf(v.w, 0.0f);
            }
            *(float4*)(sA + r * K + kv) = v;
        }
    } else {
        for (int idx = tid; idx < 16 * K; idx += 256) {
            const int r = idx / K;
            const int k = idx & (K - 1);
            const int row = m0 + r;
            float v = (row < M) ? A[(size_t)row * K + k] : 0.0f;
            if (RELU) v = fmaxf(v, 0.0f);
            sA[idx] = v;
        }
    }
    __syncthreads();

    const int wid   = tid >> 5;
    const int lane  = tid & 31;
    const int n0    = (blockIdx.y * 8 + wid) * 16;
    const int lhalf = lane >> 4;            // 0: lanes 0-15, 1: lanes 16-31
    const int lm    = lane & 15;

    const float* Bp = B + n0 + lm;          // column base for this lane
    const float* Ap = sA + lm * K + 2 * lhalf;

    v8f acc = {};
#pragma unroll 8
    for (int k0 = 0; k0 < K; k0 += 4) {
        // A fragment (ISA 16x4 f32 layout): lane -> M=lm, K = k0+2*lhalf (+1)
        v2f a = *(const v2f*)(Ap + k0);     // 8B-aligned
        // B fragment: lane -> N=n0+lm, K = k0+2*lhalf (+1)
        v2f b;
        b.x = Bp[(k0 + 2 * lhalf) * N];
        b.y = Bp[(k0 + 2 * lhalf + 1) * N];
        acc = __builtin_amdgcn_wmma_f32_16x16x4_f32(
            /*neg_a=*/false, a, /*neg_b=*/false, b,
            /*c_mod=*/(short)0, acc, /*reuse_a=*/false, /*reuse_b=*/false);
    }

    // C/D layout: VGPR r -> row m0 + r + 8*lhalf, col n0 + lm
    float* Cp = C + (size_t)(m0 + 8 * lhalf) * N + n0 + lm;
    if (full) {
#pragma unroll
        for (int r = 0; r < 8; ++r)
            Cp[r * N] = acc[r];             // immediate offsets r*N*4
    } else {
#pragma unroll
        for (int r = 0; r < 8; ++r) {
            const int row = m0 + r + 8 * lhalf;
            if (row < M) Cp[r * N] = acc[r];
        }
    }
}

// ---------------------------------------------------------------------------
// Aggregation init: out[i][j] = bias[j] + dinv[i]^2 * H[i][j]  (self-loop+bias)
// ---------------------------------------------------------------------------
template <int F>
__global__ void agg_init_kernel(const float* __restrict__ H,
                                const float* __restrict__ dinv,
                                const float* __restrict__ bias,
                                float* __restrict__ out, int n) {
    constexpr int FV = F / 4;
    size_t idx = (size_t)blockIdx.x * blockDim.x + threadIdx.x;
    if (idx >= (size_t)n * FV) return;
    const int i = (int)(idx / FV);
    const int j = (int)(idx & (FV - 1)) * 4;
    const float d2 = dinv[i] * dinv[i];
    const float4 h = *(const float4*)(H + (size_t)i * F + j);
    const float4 b = *(const float4*)(bias + j);
    float4 o;
    o.x = b.x + d2 * h.x;  o.y = b.y + d2 * h.y;
    o.z = b.z + d2 * h.z;  o.w = b.w + d2 * h.w;
    *(float4*)(out + (size_t)i * F + j) = o;
}

// ---------------------------------------------------------------------------
// Edge aggregation: out[dst] += dinv[src]*dinv[dst] * H[src]
// float4 gather (global_load_b128) + 4 hardware f32 atomics.
// ---------------------------------------------------------------------------
template <int F>
__global__ void agg_edge_kernel(const float* __restrict__ H,
                                const float* __restrict__ dinv,
                                const int* __restrict__ src,
                                const int* __restrict__ dst,
                                float* __restrict__ out, int E) {
    constexpr int TPE = F / 4;              // threads per edge (64 or 128)
    const size_t idx = (size_t)blockIdx.x * blockDim.x + threadIdx.x;
    const int e = (int)(idx / TPE);
    const int t = ((int)idx & (TPE - 1)) * 4;
    if (e >= E) return;
    const int s = src[e];
    const int d = dst[e];
    const float w = dinv[s] * dinv[d];
    const float4 h = *(const float4*)(H + (size_t)s * F + t);
    float* od = out + (size_t)d * F + t;
    atomicAdd(od + 0, w * h.x);
    atomicAdd(od + 1, w * h.y);
    atomicAdd(od + 2, w * h.z);
    atomicAdd(od + 3, w * h.w);
}

// ---------------------------------------------------------------------------
// Layer 3: per-node dot product  h3[i] = relu(A2[i,:]) . W3   (K=512, N=1)
// ---------------------------------------------------------------------------
template <int K>
__global__ void gemv_relu_kernel(const float* __restrict__ A,
                                 const float* __restrict__ w,
                                 float* __restrict__ out, int M) {
    int row  = blockIdx.x * (blockDim.x >> 5) + (threadIdx.x >> 5);
    int lane = threadIdx.x & 31;
    if (row >= M) return;
    float sum = 0.0f;
    const float* a = A + (size_t)row * K;
#pragma unroll
    for (int k = lane * 4; k < K; k += 128) {
        const float4 a4 = *(const float4*)(a + k);
        const float4 w4 = *(const float4*)(w + k);
        sum = fmaf(fmaxf(a4.x, 0.0f), w4.x, sum);
        sum = fmaf(fmaxf(a4.y, 0.0f), w4.y, sum);
        sum = fmaf(fmaxf(a4.z, 0.0f), w4.z, sum);
        sum = fmaf(fmaxf(a4.w, 0.0f), w4.w, sum);
    }
#pragma unroll
    for (int off = 16; off > 0; off >>= 1)
        sum += __shfl_down(sum, off, 32);
    if (lane == 0) out[row] = sum;
}

__global__ void scalar_init_kernel(const float* __restrict__ h3,
                                   const float* __restrict__ dinv,
                                   const float* __restrict__ b3,
                                   float* __restrict__ g, int n) {
    int i = blockIdx.x * blockDim.x + threadIdx.x;
    if (i < n) {
        float d = dinv[i];
        g[i] = b3[0] + d * d * h3[i];
    }
}

__global__ void scalar_edge_kernel(const float* __restrict__ h3,
                                   const float* __restrict__ dinv,
                                   const int* __restrict__ src,
                                   const int* __restrict__ dst,
                                   float* __restrict__ g, int e) {
    int i = blockIdx.x * blockDim.x + threadIdx.x;
    if (i < e) {
        int s = src[i], d = dst[i];
        atomicAdd(&g[d], dinv[s] * dinv[d] * h3[s]);
    }
}

__global__ void sigmoid_kernel(const float* __restrict__ g, float* __restrict__ out, int n) {
    int i = blockIdx.x * blockDim.x + threadIdx.x;
    if (i < n) out[i] = 1.0f / (1.0f + __expf(-g[i]));
}

// ---------------------------------------------------------------------------
// Launch
// ---------------------------------------------------------------------------
extern "C" void kernel_launch(void* const* d_in, const int* in_sizes, int n_in,
                              void* d_out, int out_size, void* d_ws, size_t ws_size,
                              hipStream_t stream) {
    const float* x  = (const float*)d_in[0];
    const int*   ei = (const int*)d_in[1];
    const float* W1 = (const float*)d_in[2];
    const float* b1 = (const float*)d_in[3];
    const float* W2 = (const float*)d_in[4];
    const float* b2 = (const float*)d_in[5];
    const float* W3 = (const float*)d_in[6];
    const float* b3 = (const float*)d_in[7];

    constexpr int F0 = 128, F1 = 256, F2 = 512;
    const int Nn = in_sizes[0] / F0;     // 50000
    const int E  = in_sizes[1] / 2;      // 800000
    const int* src = ei;
    const int* dst = ei + E;

    // Workspace layout (floats). A2 aliases [H1|A1] which are dead by then.
    float* ws   = (float*)d_ws;
    float* dinv = ws;                                  // Nn
    float* h3   = ws + Nn;                             // Nn
    float* g3   = ws + 2 * (size_t)Nn;                 // Nn
    size_t P0   = (3 * (size_t)Nn + 255) & ~(size_t)255;
    float* H1   = ws + P0;                             // Nn*F1
    float* A1   = H1 + (size_t)Nn * F1;                // Nn*F1
    float* H2   = ws + P0 + (size_t)Nn * F2;           // Nn*F2
    float* A2   = ws + P0;                             // Nn*F2 (aliases H1,A1)
    float* out  = (float*)d_out;

    const int T = 256;
    auto cdiv = [](size_t a, size_t b) { return (unsigned)((a + b - 1) / b); };

    // 1) symmetric normalization  dinv = rsqrt(1 + in-degree)
    deg_init_kernel<<<cdiv(Nn, T), T, 0, stream>>>(dinv, Nn);
    deg_edge_kernel<<<cdiv(E, T), T, 0, stream>>>(dst, dinv, E);
    deg_finish_kernel<<<cdiv(Nn, T), T, 0, stream>>>(dinv, Nn);

    // 2) layer 1: H1 = X @ W1 ; A1 = b1 + D^-1/2 A D^-1/2 H1
    gemm16_wmma_kernel<F0, F1, false>
        <<<dim3(cdiv(Nn, 16), F1 / 128), T, 0, stream>>>(x, W1, H1, Nn);
    agg_init_kernel<F1><<<cdiv((size_t)Nn * (F1 / 4), T), T, 0, stream>>>(H1, dinv, b1, A1, Nn);
    agg_edge_kernel<F1><<<cdiv((size_t)E * (F1 / 4), T), T, 0, stream>>>(H1, dinv, src, dst, A1, E);

    // 3) layer 2: H2 = relu(A1) @ W2 ; A2 = b2 + norm-agg(H2)
    gemm16_wmma_kernel<F1, F2, true>
        <<<dim3(cdiv(Nn, 16), F2 / 128), T, 0, stream>>>(A1, W2, H2, Nn);
    agg_init_kernel<F2><<<cdiv((size_t)Nn * (F2 / 4), T), T, 0, stream>>>(H2, dinv, b2, A2, Nn);
    agg_edge_kernel<F2><<<cdiv((size_t)E * (F2 / 4), T), T, 0, stream>>>(H2, dinv, src, dst, A2, E);

    // 4) layer 3: h3 = relu(A2) @ W3 ; g3 = b3 + norm-agg(h3) ; out = sigmoid(g3)
    gemv_relu_kernel<F2><<<cdiv(Nn, T / 32), T, 0, stream>>>(A2, W3, h3, Nn);
    scalar_init_kernel<<<cdiv(Nn, T), T, 0, stream>>>(h3, dinv, b3, g3, Nn);
    scalar_edge_kernel<<<cdiv(E, T), T, 0, stream>>>(h3, dinv, src, dst, g3, E);
    sigmoid_kernel<<<cdiv(Nn, T), T, 0, stream>>>(g3, out, Nn);
}